// VarToCon_39298950759063
// MI455X (gfx1250) — compile-verified
//
#include <hip/hip_runtime.h>

// ---------------------------------------------------------------------------
// VarToCon GNN layer, MI455X (gfx1250) plan:
//   pre_agg[c] = sum_{e: dst[e]=c} x_var[src[e]]     (fp32 HW atomics, L2-resident)
//   M    = U_h @ msg_W      (both linear layers collapsed into one 256x256)
//   cvec = U_h @ msg_b
//   out  = LN(relu( (pre_agg/(deg+eps)) @ M^T + (deg/(deg+eps))*cvec
//                   + clue*u_last + update_b ))
// GEMM: V_WMMA_F32_16X16X4_F32 (exact fp32), A tile staged to LDS per block
// via gfx1250 async LDS loads (ASYNCcnt), B streamed from L2.
// ---------------------------------------------------------------------------

#define HDIM    256
#define HP1     257
#define LN_EPS  1e-5f
#define DEG_EPS 1e-6f
#define APAD    260          // LDS row stride: dword bank = 4*l15+2*hw+k0 -> conflict-free

typedef __attribute__((ext_vector_type(2))) float v2f;
typedef __attribute__((ext_vector_type(8))) float v8f;
typedef __attribute__((ext_vector_type(4))) int   v4i;

#define AS1 __attribute__((address_space(1)))
#define AS3 __attribute__((address_space(3)))

#if defined(__has_builtin)
#if __has_builtin(__builtin_amdgcn_global_load_async_to_lds_b128)
#define HAVE_ASYNC_LDS 1
#endif
#endif

static __device__ __forceinline__ void wait_async_zero() {
#if defined(__has_builtin) && __has_builtin(__builtin_amdgcn_s_wait_asynccnt)
    __builtin_amdgcn_s_wait_asynccnt(0);
#else
    asm volatile("s_wait_asynccnt 0" ::: "memory");
#endif
}

// ---------------------------------------------------------------------------
// Kernel 1: edge scatter. One wave per edge (grid-stride). Lane l moves 8
// floats (two coalesced float4 loads) and does 8 fp32 global atomic adds.
// ---------------------------------------------------------------------------
__global__ __launch_bounds__(256) void scatter_edges(
    const float* __restrict__ x_var, const int* __restrict__ src,
    const int* __restrict__ dst, float* __restrict__ pre_agg,
    float* __restrict__ deg, int E)
{
    const int lane = threadIdx.x & 31;
    const int wid  = (blockIdx.x * (blockDim.x >> 5)) + (threadIdx.x >> 5);
    const int nw   = gridDim.x * (blockDim.x >> 5);

    for (int e = wid; e < E; e += nw) {
        const int s = src[e];
        const int d = dst[e];
        const float4* xr = (const float4*)(x_var + (size_t)s * HDIM);
        float4 v0 = xr[lane];        // cols 4*lane .. 4*lane+3
        float4 v1 = xr[lane + 32];   // cols 128+4*lane ..
        float* ar = pre_agg + (size_t)d * HDIM;
        unsafeAtomicAdd(ar + 4 * lane + 0, v0.x);
        unsafeAtomicAdd(ar + 4 * lane + 1, v0.y);
        unsafeAtomicAdd(ar + 4 * lane + 2, v0.z);
        unsafeAtomicAdd(ar + 4 * lane + 3, v0.w);
        unsafeAtomicAdd(ar + 128 + 4 * lane + 0, v1.x);
        unsafeAtomicAdd(ar + 128 + 4 * lane + 1, v1.y);
        unsafeAtomicAdd(ar + 128 + 4 * lane + 2, v1.z);
        unsafeAtomicAdd(ar + 128 + 4 * lane + 3, v1.w);
        if (lane == 0) unsafeAtomicAdd(deg + d, 1.0f);
    }
}

// ---------------------------------------------------------------------------
// Kernel 2: M = U_h @ msg_W  (256x256x256, 33 MFLOP — negligible)
// ---------------------------------------------------------------------------
__global__ __launch_bounds__(256) void combine_weights(
    const float* __restrict__ update_W, const float* __restrict__ msg_W,
    float* __restrict__ Mmat)
{
    const int i = blockIdx.x;
    const int j = threadIdx.x;
    const float* u = update_W + (size_t)i * HP1;
    float acc = 0.0f;
#pragma unroll 8
    for (int k = 0; k < HDIM; ++k)
        acc = fmaf(u[k], msg_W[(size_t)k * HDIM + j], acc);
    Mmat[(size_t)i * HDIM + j] = acc;
}

// Kernel 3: cvec = U_h @ msg_b
__global__ __launch_bounds__(256) void combine_bias(
    const float* __restrict__ update_W, const float* __restrict__ msg_b,
    float* __restrict__ cvec)
{
    const int i = threadIdx.x;
    const float* u = update_W + (size_t)i * HP1;
    float acc = 0.0f;
#pragma unroll 8
    for (int k = 0; k < HDIM; ++k) acc = fmaf(u[k], msg_b[k], acc);
    cvec[i] = acc;
}

// ---------------------------------------------------------------------------
// Kernel 4: fused GEMM + bias + ReLU + LayerNorm.
// Block = 512 threads = 16 waves; block handles 16 rows x all 256 cols.
// A tile (16x256) staged to LDS ONCE per block via async LDS loads, shared by
// all 16 waves. Wave w computes the 16x16 tile at cols [16w,16w+16) with
// 64 chained v_wmma_f32_16x16x4_f32 (K=256), A from ds_load_b64, B from L2.
//
// fp32 WMMA VGPR layouts (ISA 7.12.2):
//   A 16x4 : lane = m + 16*(k>=2), vgpr = k&1
//   B 4x16 : lane = n + 16*(k>=2), vgpr = k&1 (mirror of A); B[k][n] = M[n][k]
//   C/D    : vgpr v, lanes 0-15: (m=v, n=lane), lanes 16-31: (m=v+8, n=lane-16)
// ---------------------------------------------------------------------------
__global__ __launch_bounds__(512) void fused_gemm_ln(
    const float* __restrict__ pre_agg, const float* __restrict__ deg,
    const float* __restrict__ Mmat, const float* __restrict__ cvec,
    const float* __restrict__ clue, const float* __restrict__ update_W,
    const float* __restrict__ update_b, const float* __restrict__ ln_w,
    const float* __restrict__ ln_b, float* __restrict__ out, int num_con)
{
    alignas(16) __shared__ float As[16 * APAD];  // A tile, later reused for output staging
    __shared__ float s_df[16];                   // deg/(deg+eps) per block row
    __shared__ float s_cl[16];                   // clue per block row

    const int tid  = threadIdx.x;
    const int wave = tid >> 5;         // 0..15 -> column tile
    const int lane = tid & 31;
    const int rb   = blockIdx.x << 4;  // row tile base
    const int cb   = wave << 4;        // col tile base
    const int hw   = lane >> 4;        // lane half (selects K pair)
    const int l15  = lane & 15;

    // per-row scalars for the epilogue
    if (tid < 16) {
        int r = rb + tid; if (r >= num_con) r = num_con - 1;
        const float dg = deg[r];
        s_df[tid] = dg / (dg + DEG_EPS);
        s_cl[tid] = clue[r];
    }

    // ---- stage A tile (16 rows x 256 cols = 1024 float4) into LDS --------
#pragma unroll
    for (int it = 0; it < 2; ++it) {
        const int idx = tid + it * 512;       // float4 index
        const int row = idx >> 6;             // 64 float4 per row
        const int c4  = idx & 63;
        int gr = rb + row; if (gr >= num_con) gr = num_con - 1;
        const float* gsrc = pre_agg + (size_t)gr * HDIM + c4 * 4;
        float*       ldst = &As[row * APAD + c4 * 4];
#if defined(HAVE_ASYNC_LDS)
        __builtin_amdgcn_global_load_async_to_lds_b128(
            (AS1 v4i*)(void*)gsrc, (AS3 v4i*)(void*)ldst, 0, 0);
#else
        *(float4*)ldst = *(const float4*)gsrc;
#endif
    }
#if defined(HAVE_ASYNC_LDS)
    wait_async_zero();
#endif
    __syncthreads();

    // ---- K loop: 64 x v_wmma_f32_16x16x4_f32 ------------------------------
    int arow = rb + l15; if (arow >= num_con) arow = num_con - 1;
    const float adeg = deg[arow];
    const float invd = 1.0f / (adeg + DEG_EPS);   // folds s = pre_agg/(deg+eps)

    const float* aL   = As + l15 * APAD + (hw << 1);                  // LDS, bank-clean
    const float* bptr = Mmat + (size_t)(cb + l15) * HDIM + (hw << 1); // L2-resident

    v8f acc = {};
#pragma unroll 8
    for (int k0 = 0; k0 < HDIM; k0 += 4) {
        v2f a = *(const v2f*)(aL + k0);       // ds_load_b64
        v2f b = *(const v2f*)(bptr + k0);     // global_load_b64
        a *= invd;
        acc = __builtin_amdgcn_wmma_f32_16x16x4_f32(
            /*neg_a=*/false, a, /*neg_b=*/false, b,
            /*c_mod=*/(short)0, acc, /*reuse_a=*/false, /*reuse_b=*/false);
    }
    __syncthreads();   // all waves done reading As before it is overwritten

    // ---- epilogue: bias terms + ReLU, stage into As -----------------------
    const int   j  = cb + l15;                          // output feature
    const float cv = cvec[j];                           // (U_h@msg_b)[j]
    const float ul = update_W[(size_t)j * HP1 + HDIM];  // clue column of U
    const float ub = update_b[j];

#pragma unroll
    for (int v = 0; v < 8; ++v) {
        const int m = v + (hw << 3);          // D-tile local row
        float val = acc[v] + s_df[m] * cv + s_cl[m] * ul + ub;
        val = fmaxf(val, 0.0f);
        As[m * APAD + j] = val;               // conflict-free: bank 4m+j (mod 64)
    }
    __syncthreads();

    // ---- LayerNorm: wave w normalizes row w (wave32 shuffle reduction) ----
    const float* row = As + wave * APAD;
    float sum = 0.0f, ss = 0.0f;
#pragma unroll
    for (int c = lane; c < HDIM; c += 32) {   // stride-32: conflict-free
        const float x = row[c];
        sum += x; ss += x * x;
    }
#pragma unroll
    for (int off = 16; off > 0; off >>= 1) {
        sum += __shfl_xor(sum, off, 32);
        ss  += __shfl_xor(ss,  off, 32);
    }
    const float mu  = sum * (1.0f / HDIM);
    const float var = ss * (1.0f / HDIM) - mu * mu;
    const float rs  = rsqrtf(var + LN_EPS);

    const int orow = rb + wave;
    if (orow < num_con) {
        float4*       op  = (float4*)(out + (size_t)orow * HDIM);
        const float4* rw  = (const float4*)row;          // APAD*4 bytes is 16B-aligned
        const float4* lw4 = (const float4*)ln_w;
        const float4* lb4 = (const float4*)ln_b;
#pragma unroll
        for (int i = 0; i < 2; ++i) {         // lane writes 2 float4 = 8 floats
            const int c = lane * 2 + i;       // float4 column index 0..63
            const float4 x = rw[c];
            const float4 w = lw4[c];
            const float4 b = lb4[c];
            float4 o;
            o.x = (x.x - mu) * rs * w.x + b.x;
            o.y = (x.y - mu) * rs * w.y + b.y;
            o.z = (x.z - mu) * rs * w.z + b.z;
            o.w = (x.w - mu) * rs * w.w + b.w;
            op[c] = o;
        }
    }
}

// ---------------------------------------------------------------------------
extern "C" void kernel_launch(void* const* d_in, const int* in_sizes, int n_in,
                              void* d_out, int out_size, void* d_ws, size_t ws_size,
                              hipStream_t stream)
{
    const float* x_var    = (const float*)d_in[0];
    const int*   src      = (const int*)d_in[1];
    const int*   dst      = (const int*)d_in[2];
    const float* clue     = (const float*)d_in[3];
    // d_in[4] = num_con scalar (device); derive from in_sizes instead
    const float* msg_W    = (const float*)d_in[5];
    const float* msg_b    = (const float*)d_in[6];
    const float* update_W = (const float*)d_in[7];
    const float* update_b = (const float*)d_in[8];
    const float* ln_w     = (const float*)d_in[9];
    const float* ln_b     = (const float*)d_in[10];

    const int E       = in_sizes[1];
    const int num_con = in_sizes[3];

    // workspace layout: pre_agg | deg | M | cvec  (~51.7 MB)
    float* pre_agg = (float*)d_ws;
    float* degbuf  = pre_agg + (size_t)num_con * HDIM;
    float* Mmat    = degbuf + num_con;
    float* cvec    = Mmat + (size_t)HDIM * HDIM;

    // zero accumulators (pre_agg + deg are contiguous)
    (void)hipMemsetAsync(pre_agg, 0, (size_t)num_con * (HDIM + 1) * sizeof(float), stream);

    combine_weights<<<HDIM, 256, 0, stream>>>(update_W, msg_W, Mmat);
    combine_bias<<<1, 256, 0, stream>>>(update_W, msg_b, cvec);

    scatter_edges<<<2048, 256, 0, stream>>>(x_var, src, dst, pre_agg, degbuf, E);

    const int row_tiles = (num_con + 15) / 16;
    fused_gemm_ln<<<row_tiles, 512, 0, stream>>>(
        pre_agg, degbuf, Mmat, cvec, clue, update_W, update_b,
        ln_w, ln_b, (float*)d_out, num_con);
}